// GNNClassifier_53549652246805
// MI455X (gfx1250) — compile-verified
//
#include <hip/hip_runtime.h>
#include <hip/hip_bf16.h>

typedef __attribute__((ext_vector_type(2))) float v2f;
typedef __attribute__((ext_vector_type(8))) float v8f;

#define HDIM 128

// ---------------------------------------------------------------------------
// GEMM (proj): out[row0:row0+16, 0:128] = relu(A[16,K] @ W[K,128] + b)
// 8 waves per block; wave w computes the 16x16 tile at columns [16w,16w+16)
// A tile staged through LDS (contiguous 16*K floats in global).
// ---------------------------------------------------------------------------
__global__ __launch_bounds__(256)
void gemm128_wmma_kernel(const float* __restrict__ A, const float* __restrict__ W,
                         const float* __restrict__ bias, float* __restrict__ out,
                         int K, int do_relu)
{
    __shared__ __align__(16) float As[16 * 256];   // supports K up to 256
    const int row0 = blockIdx.x * 16;
    const int tid  = threadIdx.x;

    const float* Atile = A + (size_t)row0 * K;     // 16 x K tile, contiguous
    const int total = 16 * K;
    for (int idx = tid * 4; idx < total; idx += 256 * 4)
        *(float4*)(&As[idx]) = *(const float4*)(&Atile[idx]);
    __syncthreads();

    const int wave = tid >> 5;
    const int lane = tid & 31;
    const int m    = lane & 15;
    const int koff = (lane < 16) ? 0 : 2;          // 32-bit A 16x4 lane layout
    const int n    = wave * 16 + (lane & 15);

    v8f c = {};
    for (int k = 0; k < K; k += 4) {
        v2f a, b;
        a[0] = As[m * K + k + koff];
        a[1] = As[m * K + k + koff + 1];
        b[0] = W[(size_t)(k + koff)     * HDIM + n];
        b[1] = W[(size_t)(k + koff + 1) * HDIM + n];
        c = __builtin_amdgcn_wmma_f32_16x16x4_f32(false, a, false, b,
                                                  (short)0, c, false, false);
    }

    const int   mbase = row0 + ((lane < 16) ? 0 : 8);
    const float bv    = bias[n];
    for (int r = 0; r < 8; ++r) {
        float v = c[r] + bv;
        if (do_relu) v = fmaxf(v, 0.0f);
        out[(size_t)(mbase + r) * HDIM + n] = v;
    }
}

// ---------------------------------------------------------------------------
// Fused SAGE update: out = relu( (agg*invdeg) @ Wl + bl + h @ Wr )
// Both 128-deep K loops accumulate into the same WMMA accumulator.
// ---------------------------------------------------------------------------
__global__ __launch_bounds__(256)
void sage_update_wmma_kernel(const float* __restrict__ agg,
                             const float* __restrict__ invdeg,
                             const float* __restrict__ h,
                             const float* __restrict__ Wl,
                             const float* __restrict__ bl,
                             const float* __restrict__ Wr,
                             float* __restrict__ out)
{
    __shared__ __align__(16) float Ms[16 * HDIM];  // mean-aggregated tile
    __shared__ __align__(16) float Hs[16 * HDIM];  // self tile
    const int row0 = blockIdx.x * 16;
    const int tid  = threadIdx.x;

    for (int idx = tid * 4; idx < 16 * HDIM; idx += 256 * 4) {
        const int   r = idx / HDIM;                // constant within a float4
        const float s = invdeg[row0 + r];
        float4 va = *(const float4*)(agg + (size_t)row0 * HDIM + idx);
        va.x *= s; va.y *= s; va.z *= s; va.w *= s;
        *(float4*)(&Ms[idx]) = va;
        *(float4*)(&Hs[idx]) = *(const float4*)(h + (size_t)row0 * HDIM + idx);
    }
    __syncthreads();

    const int wave = tid >> 5;
    const int lane = tid & 31;
    const int m    = lane & 15;
    const int koff = (lane < 16) ? 0 : 2;
    const int n    = wave * 16 + (lane & 15);

    v8f c = {};
    for (int k = 0; k < HDIM; k += 4) {            // agg @ Wl
        v2f a, b;
        a[0] = Ms[m * HDIM + k + koff];
        a[1] = Ms[m * HDIM + k + koff + 1];
        b[0] = Wl[(size_t)(k + koff)     * HDIM + n];
        b[1] = Wl[(size_t)(k + koff + 1) * HDIM + n];
        c = __builtin_amdgcn_wmma_f32_16x16x4_f32(false, a, false, b,
                                                  (short)0, c, false, false);
    }
    for (int k = 0; k < HDIM; k += 4) {            // + h @ Wr
        v2f a, b;
        a[0] = Hs[m * HDIM + k + koff];
        a[1] = Hs[m * HDIM + k + koff + 1];
        b[0] = Wr[(size_t)(k + koff)     * HDIM + n];
        b[1] = Wr[(size_t)(k + koff + 1) * HDIM + n];
        c = __builtin_amdgcn_wmma_f32_16x16x4_f32(false, a, false, b,
                                                  (short)0, c, false, false);
    }

    const int   mbase = row0 + ((lane < 16) ? 0 : 8);
    const float bv    = bl[n];
    for (int r = 0; r < 8; ++r) {
        float v = fmaxf(c[r] + bv, 0.0f);
        out[(size_t)(mbase + r) * HDIM + n] = v;
    }
}

// ---------------------------------------------------------------------------
// Classifier: logits[row0:row0+16, tile] = h @ W_cls + b_cls  (ncls=40, 3 tiles)
// One wave per 16x16 tile; columns >= ncls masked (loads 0, stores skipped).
// ---------------------------------------------------------------------------
__global__ __launch_bounds__(32)
void cls_wmma_kernel(const float* __restrict__ h, const float* __restrict__ Wc,
                     const float* __restrict__ bc, float* __restrict__ out, int ncls)
{
    const int row0 = blockIdx.x * 16;
    const int lane = threadIdx.x & 31;
    const int m    = lane & 15;
    const int koff = (lane < 16) ? 0 : 2;
    const int n    = blockIdx.y * 16 + (lane & 15);
    const bool nv  = (n < ncls);

    v8f c = {};
    for (int k = 0; k < HDIM; k += 4) {
        v2f a, b;
        a[0] = h[(size_t)(row0 + m) * HDIM + k + koff];
        a[1] = h[(size_t)(row0 + m) * HDIM + k + koff + 1];
        b[0] = nv ? Wc[(size_t)(k + koff)     * ncls + n] : 0.0f;  // select, no EXEC change
        b[1] = nv ? Wc[(size_t)(k + koff + 1) * ncls + n] : 0.0f;
        c = __builtin_amdgcn_wmma_f32_16x16x4_f32(false, a, false, b,
                                                  (short)0, c, false, false);
    }

    if (nv) {
        const int   mbase = row0 + ((lane < 16) ? 0 : 8);
        const float bv    = bc[n];
        for (int r = 0; r < 8; ++r)
            out[(size_t)(mbase + r) * ncls + n] = c[r] + bv;
    }
}

// ---------------------------------------------------------------------------
// Edge scatter: one wave per edge, each lane carries a float4 (32*4 = 128 ch).
// ---------------------------------------------------------------------------
__global__ __launch_bounds__(256)
void scatter_add_kernel(const int* __restrict__ src, const int* __restrict__ dst,
                        const float* __restrict__ h, float* __restrict__ agg, int E)
{
    const int e = blockIdx.x * 8 + (threadIdx.x >> 5);
    if (e >= E) return;
    const int lane = threadIdx.x & 31;
    const int s = src[e], d = dst[e];
    float4 v = *(const float4*)(h + (size_t)s * HDIM + lane * 4);
    float* a = agg + (size_t)d * HDIM + lane * 4;
    atomicAdd(a + 0, v.x);
    atomicAdd(a + 1, v.y);
    atomicAdd(a + 2, v.z);
    atomicAdd(a + 3, v.w);
}

__global__ __launch_bounds__(256)
void deg_kernel(const int* __restrict__ dst, float* __restrict__ deg, int E)
{
    const int e = blockIdx.x * blockDim.x + threadIdx.x;
    if (e < E) atomicAdd(&deg[dst[e]], 1.0f);
}

__global__ __launch_bounds__(256)
void invdeg_kernel(float* __restrict__ deg, int n)
{
    const int i = blockIdx.x * blockDim.x + threadIdx.x;
    if (i < n) deg[i] = 1.0f / fmaxf(deg[i], 1.0f);
}

// ---------------------------------------------------------------------------
extern "C" void kernel_launch(void* const* d_in, const int* in_sizes, int n_in,
                              void* d_out, int out_size, void* d_ws, size_t ws_size,
                              hipStream_t stream)
{
    const float* x      = (const float*)d_in[0];
    const int*   edge   = (const int*)  d_in[1];
    const float* W_proj = (const float*)d_in[2];
    const float* b_proj = (const float*)d_in[3];
    const float* Wl[3]  = { (const float*)d_in[4], (const float*)d_in[7], (const float*)d_in[10] };
    const float* bl[3]  = { (const float*)d_in[5], (const float*)d_in[8], (const float*)d_in[11] };
    const float* Wr[3]  = { (const float*)d_in[6], (const float*)d_in[9], (const float*)d_in[12] };
    const float* Wc     = (const float*)d_in[13];
    const float* bc     = (const float*)d_in[14];

    const int N    = in_sizes[0] / 256;    // IN_FT = 256
    const int E    = in_sizes[1] / 2;      // edge_index is [2, E]
    const int ncls = in_sizes[14];         // 40

    const int* src = edge;
    const int* dst = edge + E;

    // Workspace: hA | hB | agg | invdeg   (3*N*H + N floats ~ 154 MB)
    float* hA     = (float*)d_ws;
    float* hB     = hA  + (size_t)N * HDIM;
    float* agg    = hB  + (size_t)N * HDIM;
    float* invdeg = agg + (size_t)N * HDIM;

    const int rowTiles = (N + 15) / 16;    // 6250 (N % 16 == 0)

    // Degrees -> 1/max(deg,1), computed once, reused for all three layers.
    hipMemsetAsync(invdeg, 0, (size_t)N * sizeof(float), stream);
    deg_kernel<<<(E + 255) / 256, 256, 0, stream>>>(dst, invdeg, E);
    invdeg_kernel<<<(N + 255) / 256, 256, 0, stream>>>(invdeg, N);

    // Projection + ReLU
    gemm128_wmma_kernel<<<rowTiles, 256, 0, stream>>>(x, W_proj, b_proj, hA, 256, 1);

    // Three SAGE layers
    for (int l = 0; l < 3; ++l) {
        hipMemsetAsync(agg, 0, (size_t)N * HDIM * sizeof(float), stream);
        scatter_add_kernel<<<(E + 7) / 8, 256, 0, stream>>>(src, dst, hA, agg, E);
        sage_update_wmma_kernel<<<rowTiles, 256, 0, stream>>>(agg, invdeg, hA,
                                                              Wl[l], bl[l], Wr[l], hB);
        float* t = hA; hA = hB; hB = t;
    }

    // Classifier
    cls_wmma_kernel<<<dim3(rowTiles, (ncls + 15) / 16), 32, 0, stream>>>(
        hA, Wc, bc, (float*)d_out, ncls);
}